// CTC_55808805045003
// MI455X (gfx1250) — compile-verified
//
#include <hip/hip_runtime.h>
#include <stdint.h>

// Problem constants (from reference setup_inputs):
//   B = 1024, T = 256, C = 128, L = 64  ->  S = 2L+1 = 129, blank = C-1
#define T_STEPS   256
#define C_CLASSES 128
#define L_LABELS  64
#define BLANK_ID  (C_CLASSES - 1)      // 127
#define NBUF      16                   // ring-buffer depth (rows in flight), power of 2
#define NEGF      (-1.0e30f)
#define EPSF      (1.0e-7f)
#define NTHREADS  128                  // 4 waves; thread s owns state s, thread 127
                                       // additionally owns state 128 in registers

// ---------------------------------------------------------------------------
// CDNA5 async global->LDS copy of one 512B probability row.
// 32 lanes x b128 (16B each) = 512B = one C_CLASSES*f32 row, ONE async op,
// tracked by ASYNCcnt. Inline asm (portable across ROCm7.2 / amdgpu-toolchain).
// the LDS destination address comes from the VDST VGPR: ptrtoint of an addrspace(3) pointer gives
// exactly the wave-relative LDS byte offset the instruction wants.
// ---------------------------------------------------------------------------
__device__ __forceinline__ void async_copy_row(const float* __restrict__ gsrc,
                                               float* ldst, int lane) {
  unsigned lds_off = (unsigned)(size_t)(__attribute__((address_space(3))) float*)(ldst + lane * 4);
  const float* gaddr = gsrc + lane * 4;
  asm volatile("global_load_async_to_lds_b128 %0, %1, off"
               :: "v"(lds_off), "v"(gaddr)
               : "memory");
}

// Steady-state wait: with the "issue row t+NBUF-1 at step t" schedule,
// row t is guaranteed complete once outstanding <= NBUF-2.
__device__ __forceinline__ void wait_async_15() { asm volatile("s_wait_asynccnt 15" ::: "memory"); }
__device__ __forceinline__ void wait_async_14() { asm volatile("s_wait_asynccnt 14" ::: "memory"); }
__device__ __forceinline__ void wait_async_0()  { asm volatile("s_wait_asynccnt 0"  ::: "memory"); }

// log(exp(a)+exp(b)+exp(c)) with -1e30 sentinels (safe: the max term is exp(0)=1)
__device__ __forceinline__ float lse3(float a, float b, float c) {
  float m = fmaxf(a, fmaxf(b, c));
  return m + __logf(__expf(a - m) + __expf(b - m) + __expf(c - m));
}
__device__ __forceinline__ float lse2(float a, float b) {
  float m = fmaxf(a, b);
  return m + __logf(__expf(a - m) + __expf(b - m));
}

__global__ __launch_bounds__(NTHREADS)
void ctc_loss_kernel(const int* __restrict__ labels,    // [B, L]
                     const float* __restrict__ probs,   // [B, T, C] softmax
                     float* __restrict__ out) {         // [B, 1]
  __shared__ float rows[NBUF][C_CLASSES];   // ring buffer of prob rows (8 KB)
  __shared__ float albuf[2][C_CLASSES];     // ping-pong alpha, states 0..127 (1 KB)

  const int b   = blockIdx.x;
  const int tid = threadIdx.x;
  const int s   = tid;                      // extended-label state index
  const float* pb = probs + (size_t)b * T_STEPS * C_CLASSES;

  // ---- prologue: wave 0 launches rows 0..NBUF-1 into the ring buffer ----
  if (tid < 32) {
#pragma unroll
    for (int r = 0; r < NBUF; ++r)
      async_copy_row(pb + r * C_CLASSES, &rows[r][0], tid);
  }

  // ---- per-state setup: symbol + skip-transition legality ----
  int myclass = BLANK_ID;
  int skipok  = 0;
  if (s & 1) {                         // odd state -> label (s-1)/2
    int li  = s >> 1;
    myclass = labels[b * L_LABELS + li];
    // skip (s-2 -> s) legal iff symbol != blank (always: labels < C-1) and
    // symbol != previous label; state 1's skip source is the NEG pad -> 0.
    skipok  = (li > 0) && (myclass != labels[b * L_LABELS + li - 1]);
  }
  const bool last = (tid == NTHREADS - 1);

  // ---- t = 0 init: only states 0 (blank) and 1 (first label) reachable ----
  if (tid < 32) wait_async_15();       // >=1 completed => row 0 landed
  __syncthreads();
  float a  = (s <= 1) ? __logf(rows[0][myclass] + EPSF) : NEGF;  // alpha_0[s]
  float a2 = NEGF;                     // hidden state 128 (thread 127 only)
  albuf[0][s] = a;

  // ---- main DP: ONE barrier per timestep via alpha ping-pong ----
  for (int t = 1; t < T_STEPS; ++t) {
    if (tid < 32) {
      // issued before this wait: min(T, t+NBUF-1) rows; need rows 0..t done
      if (t <= T_STEPS - NBUF + 1) wait_async_14();
      else                         wait_async_0();
    }
    __syncthreads();  // publishes: row t in LDS, albuf[(t-1)&1] complete,
                      // and ring slot (t-1)&(NBUF-1) fully consumed

    if (tid < 32) {                    // refill the slot freed by step t-1
      int r = t + NBUF - 1;
      if (r < T_STEPS)
        async_copy_row(pb + (size_t)r * C_CLASSES, &rows[r & (NBUF - 1)][0], tid);
    }

    const float* al = albuf[(t - 1) & 1];
    float am1 = (s >= 1) ? al[s - 1] : NEGF;
    float am2 = skipok   ? al[s - 2] : NEGF;         // skipok => s >= 3
    float lp  = __logf(rows[t & (NBUF - 1)][myclass] + EPSF);
    float anew = lse3(a, am1, am2) + lp;

    if (last) {                        // state 128: blank, depends on 128 & 127 only
      float lpb = __logf(rows[t & (NBUF - 1)][BLANK_ID] + EPSF);
      a2 = lse2(a2, a) + lpb;          // uses old alpha_{t-1}[127] (register)
    }
    a = anew;
    albuf[t & 1][s] = anew;            // other buffer than the one being read
  }

  // ---- finish: -logaddexp(alpha_T[128], alpha_T[127]) from registers ----
  if (last) out[b] = -lse2(a2, a);
}

extern "C" void kernel_launch(void* const* d_in, const int* in_sizes, int n_in,
                              void* d_out, int out_size, void* d_ws, size_t ws_size,
                              hipStream_t stream) {
  const int*   y_true = (const int*)d_in[0];    // [B, L] int32
  const float* y_pred = (const float*)d_in[1];  // [B, T, C] float32
  float*       outp   = (float*)d_out;          // [B, 1] float32

  const int B = in_sizes[0] / L_LABELS;         // 1024
  ctc_loss_kernel<<<dim3(B), dim3(NTHREADS), 0, stream>>>(y_true, y_pred, outp);
}